// SoftMoELayerWrapper_5042291606066
// MI455X (gfx1250) — compile-verified
//
#include <hip/hip_runtime.h>
#include <hip/hip_bf16.h>
#include <math.h>

// ---------------------------------------------------------------------------
// Soft-MoE (B=8, M=4096, D=512, E=64, P=1, H=2048) for MI455X (gfx1250).
// Memory-bound (~0.7 GB traffic vs ~10.7 GFLOP) -> exact-precision
// V_WMMA_F32_16X16X4_F32 for all five GEMMs; LDS staging uses the CDNA5
// async-to-LDS path (GLOBAL_LOAD_ASYNC_TO_LDS_B128 + s_wait_asynccnt),
// normalization scales are fused into fragment builds (xn never hits HBM).
// ---------------------------------------------------------------------------

typedef __attribute__((ext_vector_type(2))) float v2f;
typedef __attribute__((ext_vector_type(4))) float v4f;
typedef __attribute__((ext_vector_type(8))) float v8f;

#define NB 8
#define NM 4096
#define ND 512
#define NE 64
#define NH 2048

static __device__ __forceinline__ v8f wmma4(v2f a, v2f b, v8f c) {
  // V_WMMA_F32_16X16X4_F32: D(16x16 f32) = A(16x4 f32) * B(4x16 f32) + C
  return __builtin_amdgcn_wmma_f32_16x16x4_f32(false, a, false, b, (short)0, c,
                                               false, false);
}
static __device__ __forceinline__ v8f zero8() {
  v8f z = {0.f, 0.f, 0.f, 0.f, 0.f, 0.f, 0.f, 0.f};
  return z;
}
static __device__ __forceinline__ float gelu_exact(float x) {
  return 0.5f * x * (1.0f + erff(x * 0.70710678118654752440f));
}

// CDNA5 async copy: 16 bytes per lane, global -> LDS, tracked by ASYNCcnt.
// GVS addressing: global address is the 64-bit SGPR base plus the 32-bit VADDR offset; LDS address is in VDST.
// Low 32 bits of a generic LDS pointer are the raw LDS byte address
// (flat-aperture scheme, ISA 10.2), so a truncating cast gives VDST.
static __device__ __forceinline__ void async_ld_b128(const float* gbase,
                                                     unsigned gbyteoff,
                                                     const float* ldsptr) {
  unsigned lds = (unsigned)(unsigned long long)ldsptr;
  asm volatile("global_load_async_to_lds_b128 %0, %1, %2"
               :
               : "v"(lds), "v"(gbyteoff), "s"(gbase)
               : "memory");
}
static __device__ __forceinline__ void async_wait0() {
  asm volatile("s_wait_asynccnt 0x0" ::: "memory");
}

// ---------------------------------------------------------------------------
// Kernel 1: rn[row] = 1/max(||x_row||, eps).  grid 4096, block 256.
// ---------------------------------------------------------------------------
__global__ void k_rownorm(const float* __restrict__ x, float* __restrict__ rn) {
  const int wave = threadIdx.x >> 5, lane = threadIdx.x & 31;
  const size_t row = (size_t)blockIdx.x * 8 + wave;  // 0..32767
  const float* xr = x + row * ND;
  float ss = 0.f;
#pragma unroll
  for (int j = 0; j < 4; ++j) {
    v4f v = reinterpret_cast<const v4f*>(xr)[lane + 32 * j];
    ss += v[0] * v[0] + v[1] * v[1] + v[2] * v[2] + v[3] * v[3];
  }
#pragma unroll
  for (int off = 16; off >= 1; off >>= 1) ss += __shfl_xor(ss, off, 32);
  if (lane == 0) rn[row] = 1.0f / fmaxf(sqrtf(ss), 1e-12f);
}

// ---------------------------------------------------------------------------
// Kernel 2: phin[d][e] = scale * phi[d][e]/max(||phi[:,e]||,eps). 1x64.
// ---------------------------------------------------------------------------
__global__ void k_norm_phi(const float* __restrict__ phi,
                           const float* __restrict__ scale,
                           float* __restrict__ phin) {
  const int e = threadIdx.x;  // 0..63
  float ss = 0.f;
  for (int d = 0; d < ND; ++d) {
    float p = phi[(size_t)d * NE + e];
    ss += p * p;
  }
  float r = scale[0] / fmaxf(sqrtf(ss), 1e-12f);
  for (int d = 0; d < ND; ++d)
    phin[(size_t)d * NE + e] = phi[(size_t)d * NE + e] * r;
}

// ---------------------------------------------------------------------------
// Kernel 3: logits = (x*rn) @ phin   (32768 x 64 x 512).
// Wave: 16 rows x 64 cols; phin (128 KB) is L2-resident.  grid 256.
// ---------------------------------------------------------------------------
__global__ void k_logits(const float* __restrict__ x,
                         const float* __restrict__ rn,
                         const float* __restrict__ phin,
                         float* __restrict__ logits) {
  const int wave = threadIdx.x >> 5, lane = threadIdx.x & 31;
  const int half = lane >> 4, l16 = lane & 15;
  const size_t row0 = ((size_t)blockIdx.x * 8 + wave) * 16;
  const float r = rn[row0 + l16];
  const float* arow = x + (row0 + l16) * ND + 2 * half;
  const float* bbase = phin + (size_t)(2 * half) * NE + l16;
  v8f acc[4];
#pragma unroll
  for (int ct = 0; ct < 4; ++ct) acc[ct] = zero8();
#pragma unroll 4
  for (int k = 0; k < ND; k += 4) {
    v2f a = *reinterpret_cast<const v2f*>(arow + k);
    a[0] *= r;
    a[1] *= r;
    const float* bp = bbase + (size_t)k * NE;
#pragma unroll
    for (int ct = 0; ct < 4; ++ct) {
      v2f b;
      b[0] = bp[ct * 16];
      b[1] = bp[NE + ct * 16];
      acc[ct] = wmma4(a, b, acc[ct]);
    }
  }
#pragma unroll
  for (int ct = 0; ct < 4; ++ct)
#pragma unroll
    for (int i = 0; i < 8; ++i)
      logits[(row0 + i + 8 * half) * NE + ct * 16 + l16] = acc[ct][i];
}

// ---------------------------------------------------------------------------
// Kernel 4: dispatch softmax over tokens m -> dT[b][e][m].  grid 512.
// ---------------------------------------------------------------------------
__global__ void k_softmax_m(const float* __restrict__ logits,
                            float* __restrict__ dT) {
  __shared__ float red[256];
  const int b = blockIdx.x >> 6, e = blockIdx.x & 63, tid = threadIdx.x;
  const float* src = logits + (size_t)b * NM * NE + e;
  float mx = -3.4e38f;
  for (int m = tid; m < NM; m += 256) mx = fmaxf(mx, src[(size_t)m * NE]);
  red[tid] = mx;
  __syncthreads();
  for (int s = 128; s >= 1; s >>= 1) {
    if (tid < s) red[tid] = fmaxf(red[tid], red[tid + s]);
    __syncthreads();
  }
  mx = red[0];
  __syncthreads();
  float sum = 0.f;
  for (int m = tid; m < NM; m += 256) sum += expf(src[(size_t)m * NE] - mx);
  red[tid] = sum;
  __syncthreads();
  for (int s = 128; s >= 1; s >>= 1) {
    if (tid < s) red[tid] += red[tid + s];
    __syncthreads();
  }
  const float rs = 1.0f / red[0];
  float* dst = dT + ((size_t)b * NE + e) * NM;
  for (int m = tid; m < NM; m += 256)
    dst[m] = expf(src[(size_t)m * NE] - mx) * rs;
}

// ---------------------------------------------------------------------------
// Kernel 5: combine softmax over experts (row of 64), in-place.  grid 4096.
// ---------------------------------------------------------------------------
__global__ void k_softmax_e(float* __restrict__ cbuf) {
  const int wave = threadIdx.x >> 5, lane = threadIdx.x & 31;
  const size_t row = (size_t)blockIdx.x * 8 + wave;
  float* p = cbuf + row * NE + lane;
  float v0 = p[0], v1 = p[32];
  float mx = fmaxf(v0, v1);
#pragma unroll
  for (int off = 16; off >= 1; off >>= 1)
    mx = fmaxf(mx, __shfl_xor(mx, off, 32));
  float e0 = expf(v0 - mx), e1 = expf(v1 - mx);
  float s = e0 + e1;
#pragma unroll
  for (int off = 16; off >= 1; off >>= 1) s += __shfl_xor(s, off, 32);
  const float rs = 1.0f / s;
  p[0] = e0 * rs;
  p[32] = e1 * rs;
}

// ---------------------------------------------------------------------------
// Kernel 6: dispatch GEMM  xs_part[ks][b] = (dT[b]*rn) @ x[b]  over a K-slice.
// x and dT chunks staged to LDS with async b128 copies (no VGPR round-trip);
// rn applied to A-fragments via a 16-entry LDS broadcast table.
// grid (8 batches, 8 K-slices), block 256 (8 waves; wave owns 64x64 out).
// ---------------------------------------------------------------------------
__global__ void k_dispatch(const float* __restrict__ dT,
                           const float* __restrict__ x,
                           const float* __restrict__ rn,
                           float* __restrict__ xsp) {
  __shared__ __attribute__((aligned(16))) float s_xn[16 * ND];  // 32 KB
  __shared__ __attribute__((aligned(16))) float s_dT[NE * 20];  // pad 20
  __shared__ float s_rnc[16];
  const int b = blockIdx.x, ks = blockIdx.y;
  const int tid = threadIdx.x;
  const int wave = tid >> 5, lane = tid & 31, half = lane >> 4, l16 = lane & 15;
  const int col0 = wave * 64;
  const float* xb = x + (size_t)b * NM * ND;        // uniform sgpr base
  const float* dTb = dT + (size_t)b * NE * NM;      // uniform sgpr base
  v8f acc[4][4];
#pragma unroll
  for (int rt = 0; rt < 4; ++rt)
#pragma unroll
    for (int ct = 0; ct < 4; ++ct) acc[rt][ct] = zero8();

  const int kc0 = ks * 512;
  for (int kc = kc0; kc < kc0 + 512; kc += 16) {
    __syncthreads();  // previous iteration done reading LDS
    // ---- async stage 16x512 chunk of x (contiguous 32 KB)
#pragma unroll
    for (int j = 0; j < 8; ++j) {
      int idx = tid + 256 * j;  // 16-byte units, 2048 total
      async_ld_b128(xb, (unsigned)(kc * ND * 4) + (unsigned)idx * 16u,
                    s_xn + idx * 4);
    }
    // ---- async stage 64x16 chunk of dT (16B per thread)
    {
      int e = tid >> 2, kk4 = (tid & 3) * 4;
      async_ld_b128(dTb, (unsigned)((e * NM + kc + kk4) * 4),
                    s_dT + e * 20 + kk4);
    }
    if (tid < 16) s_rnc[tid] = rn[(size_t)b * NM + kc + tid];
    async_wait0();   // our own async copies complete
    __syncthreads(); // everyone's copies complete
#pragma unroll
    for (int kk = 0; kk < 16; kk += 4) {
      const float r0 = s_rnc[kk + 2 * half];
      const float r1 = s_rnc[kk + 2 * half + 1];
      v2f afr[4];
#pragma unroll
      for (int rt = 0; rt < 4; ++rt) {
        v2f t = *reinterpret_cast<const v2f*>(
            &s_dT[(rt * 16 + l16) * 20 + kk + 2 * half]);
        t[0] *= r0;
        t[1] *= r1;
        afr[rt] = t;
      }
#pragma unroll
      for (int ct = 0; ct < 4; ++ct) {
        v2f bfr;
        bfr[0] = s_xn[(kk + 2 * half) * ND + col0 + ct * 16 + l16];
        bfr[1] = s_xn[(kk + 2 * half + 1) * ND + col0 + ct * 16 + l16];
#pragma unroll
        for (int rt = 0; rt < 4; ++rt)
          acc[rt][ct] = wmma4(afr[rt], bfr, acc[rt][ct]);
      }
    }
  }
  // xsp layout [KS][B][E][D]
  float* out = xsp + (((size_t)ks * NB + b) * NE) * ND;
#pragma unroll
  for (int rt = 0; rt < 4; ++rt)
#pragma unroll
    for (int ct = 0; ct < 4; ++ct)
#pragma unroll
      for (int i = 0; i < 8; ++i)
        out[(size_t)(rt * 16 + i + 8 * half) * ND + col0 + ct * 16 + l16] =
            acc[rt][ct][i];
}

// grid 1024, block 256: xs = sum over 8 K-slices
__global__ void k_reduce_xs(const float* __restrict__ xsp,
                            float* __restrict__ xs) {
  size_t i = (size_t)blockIdx.x * 256 + threadIdx.x;  // 262144 elements
  float s = 0.f;
#pragma unroll
  for (int k = 0; k < 8; ++k) s += xsp[(size_t)k * (NB * NE * ND) + i];
  xs[i] = s;
}

// ---------------------------------------------------------------------------
// Kernel 7: expert MLP per (expert e, H-slice hs) block:
//   A (8 slots padded to 16 rows) async-staged into LDS;
//   GEMM1 -> +b1 -> exact GELU -> h-tile in LDS -> fused GEMM2 partial.
// Weights (the 512 MB stream) are read once, coalesced, with prefetch.
// grid (64, 4), block 256.
// ---------------------------------------------------------------------------
__global__ void k_expert_mlp(const float* __restrict__ xs,
                             const float* __restrict__ w1,
                             const float* __restrict__ b1,
                             const float* __restrict__ w2,
                             float* __restrict__ ysp) {
  __shared__ __attribute__((aligned(16))) float sA[16 * 520];  // ~33 KB
  __shared__ __attribute__((aligned(16))) float sH[16 * 132];  // ~8 KB
  const int e = blockIdx.x, hs = blockIdx.y;
  const int tid = threadIdx.x;
  const int wave = tid >> 5, lane = tid & 31, half = lane >> 4, l16 = lane & 15;

  // zero padding region (rows 8..15 fully, pad cols of rows 0..7): disjoint
  // from the async destination, so DS stores and async LDS writes can't race.
  for (int i = tid; i < 8 * 520; i += 256) sA[8 * 520 + i] = 0.f;
  if (tid < 64) sA[(tid >> 3) * 520 + 512 + (tid & 7)] = 0.f;
  // async-stage rows 0..7: xs[bb][e][:] (16B per copy)
  const float* xse = xs + (size_t)e * ND;  // uniform sgpr base
#pragma unroll
  for (int j = 0; j < 4; ++j) {
    int i4 = tid + 256 * j;  // 1024 x 16B
    int bb = i4 >> 7, d4 = i4 & 127;
    async_ld_b128(xse, (unsigned)(bb * NE * ND + d4 * 4) * 4u,
                  sA + bb * 520 + d4 * 4);
  }
  async_wait0();
  __syncthreads();

  v8f acc2[4];
#pragma unroll
  for (int ct = 0; ct < 4; ++ct) acc2[ct] = zero8();

  const int hbase = hs * 512;
  const float* w1e = w1 + (size_t)e * ND * NH;
  const float* w2e = w2 + (size_t)e * NH * ND;
  const float* sAr = sA + l16 * 520 + 2 * half;

  for (int hc = 0; hc < 512; hc += 128) {
    // ---- GEMM1: this wave's 16x16 h-tile at absolute column h0
    const int h0 = hbase + hc + wave * 16;
    v8f acc1 = zero8();
    const float* w1p = w1e + (size_t)(2 * half) * NH + h0 + l16;
    __builtin_prefetch(w1p + (size_t)64 * NH, 0, 1);  // stream hint
#pragma unroll 4
    for (int k = 0; k < ND; k += 4) {
      v2f a = *reinterpret_cast<const v2f*>(sAr + k);
      v2f bv;
      bv[0] = w1p[(size_t)k * NH];
      bv[1] = w1p[(size_t)(k + 1) * NH];
      acc1 = wmma4(a, bv, acc1);
    }
    const float bias = b1[(size_t)e * NH + h0 + l16];
    __syncthreads();  // previous GEMM2 finished reading sH
#pragma unroll
    for (int i = 0; i < 8; ++i) {
      float hv = gelu_exact(acc1[i] + bias);
      sH[(i + 8 * half) * 132 + wave * 16 + l16] = hv;
    }
    __syncthreads();
    // ---- GEMM2 partial over this 128-wide K slice; wave owns 64 ys columns
    const float* sHr = sH + l16 * 132 + 2 * half;
    const float* w2p =
        w2e + (size_t)(hbase + hc + 2 * half) * ND + wave * 64 + l16;
    __builtin_prefetch(w2p + (size_t)64 * ND, 0, 1);  // stream hint
#pragma unroll 2
    for (int k2 = 0; k2 < 128; k2 += 4) {
      v2f a2 = *reinterpret_cast<const v2f*>(sHr + k2);
#pragma unroll
      for (int ct = 0; ct < 4; ++ct) {
        v2f bv;
        bv[0] = w2p[(size_t)k2 * ND + ct * 16];
        bv[1] = w2p[(size_t)(k2 + 1) * ND + ct * 16];
        acc2[ct] = wmma4(a2, bv, acc2[ct]);
      }
    }
  }
  // ysp layout [HS][B][E][D]; rows 8..15 were padding -> dropped
#pragma unroll
  for (int i = 0; i < 8; ++i) {
    int r = i + 8 * half;
    if (r < 8) {
      float* out =
          ysp + (((size_t)hs * NB + r) * NE + e) * ND + wave * 64 + l16;
#pragma unroll
      for (int ct = 0; ct < 4; ++ct) out[ct * 16] = acc2[ct][i];
    }
  }
}

// grid 1024, block 256: ys = sum over 4 H-slices + b2
__global__ void k_reduce_ys(const float* __restrict__ ysp,
                            const float* __restrict__ b2,
                            float* __restrict__ ys) {
  size_t i = (size_t)blockIdx.x * 256 + threadIdx.x;  // 262144 elements
  float s = b2[i & (NE * ND - 1)];
#pragma unroll
  for (int k = 0; k < 4; ++k) s += ysp[(size_t)k * (NB * NE * ND) + i];
  ys[i] = s;
}

// ---------------------------------------------------------------------------
// Kernel 8: combine  y[b] = c[b](4096x64) @ ys[b](64x512); ys L2-resident.
// grid 2048 (b * 256 m-tiles), block 256 (8 waves x 64 cols each).
// ---------------------------------------------------------------------------
__global__ void k_combine(const float* __restrict__ cmat,
                          const float* __restrict__ ys,
                          float* __restrict__ y) {
  const int b = blockIdx.x >> 8, mt = blockIdx.x & 255;
  const int tid = threadIdx.x;
  const int wave = tid >> 5, lane = tid & 31, half = lane >> 4, l16 = lane & 15;
  const int m0 = mt * 16, col0 = wave * 64;
  v8f acc[4];
#pragma unroll
  for (int ct = 0; ct < 4; ++ct) acc[ct] = zero8();
  const float* ar = cmat + ((size_t)b * NM + m0 + l16) * NE + 2 * half;
  const float* br =
      ys + (size_t)b * NE * ND + (size_t)(2 * half) * ND + col0 + l16;
#pragma unroll
  for (int k = 0; k < NE; k += 4) {
    v2f a = *reinterpret_cast<const v2f*>(ar + k);
#pragma unroll
    for (int ct = 0; ct < 4; ++ct) {
      v2f bv;
      bv[0] = br[(size_t)k * ND + ct * 16];
      bv[1] = br[(size_t)(k + 1) * ND + ct * 16];
      acc[ct] = wmma4(a, bv, acc[ct]);
    }
  }
  float* out = y + ((size_t)b * NM + m0 + 8 * half) * ND + col0 + l16;
#pragma unroll
  for (int i = 0; i < 8; ++i)
#pragma unroll
    for (int ct = 0; ct < 4; ++ct)
      out[(size_t)i * ND + ct * 16] = acc[ct][i];
}

// ---------------------------------------------------------------------------
extern "C" void kernel_launch(void* const* d_in, const int* in_sizes, int n_in,
                              void* d_out, int out_size, void* d_ws,
                              size_t ws_size, hipStream_t stream) {
  (void)in_sizes; (void)n_in; (void)out_size; (void)ws_size;
  const float* x     = (const float*)d_in[0];  // [B,M,D]
  const float* phi   = (const float*)d_in[1];  // [D,E]
  const float* scale = (const float*)d_in[2];  // [1]
  const float* w1    = (const float*)d_in[3];  // [E,D,H]
  const float* b1    = (const float*)d_in[4];  // [E,H]
  const float* w2    = (const float*)d_in[5];  // [E,H,D]
  const float* b2    = (const float*)d_in[6];  // [E,D]
  float* y = (float*)d_out;                    // [B,M,D]

  // workspace carve-up (floats); total ~ 31.7 MB
  float* ws     = (float*)d_ws;
  float* rn     = ws;                                   // 32768
  float* phin   = rn + (size_t)NB * NM;                 // 32768
  float* logits = phin + (size_t)ND * NE;               // 2,097,152 (also c)
  float* dT     = logits + (size_t)NB * NM * NE;        // 2,097,152
  float* xsp    = dT + (size_t)NB * NE * NM;            // 8 * 262,144
  float* xsbuf  = xsp + (size_t)8 * NB * NE * ND;       // 262,144
  float* ysp    = xsbuf + (size_t)NB * NE * ND;         // 4 * 262,144
  float* ysbuf  = ysp + (size_t)4 * NB * NE * ND;       // 262,144

  k_rownorm<<<dim3((NB * NM) / 8), dim3(256), 0, stream>>>(x, rn);
  k_norm_phi<<<dim3(1), dim3(64), 0, stream>>>(phi, scale, phin);
  k_logits<<<dim3((NB * NM) / 128), dim3(256), 0, stream>>>(x, rn, phin,
                                                            logits);
  k_softmax_m<<<dim3(NB * NE), dim3(256), 0, stream>>>(logits, dT);
  k_softmax_e<<<dim3((NB * NM) / 8), dim3(256), 0, stream>>>(logits);
  k_dispatch<<<dim3(NB, 8), dim3(256), 0, stream>>>(dT, x, rn, xsp);
  k_reduce_xs<<<dim3((NB * NE * ND) / 256), dim3(256), 0, stream>>>(xsp,
                                                                    xsbuf);
  k_expert_mlp<<<dim3(NE, 4), dim3(256), 0, stream>>>(xsbuf, w1, b1, w2, ysp);
  k_reduce_ys<<<dim3((NB * NE * ND) / 256), dim3(256), 0, stream>>>(ysp, b2,
                                                                    ysbuf);
  k_combine<<<dim3(NB * (NM / 16)), dim3(256), 0, stream>>>(logits, ysbuf, y);
}